// AGCRNCellWithMLP_59253368815847
// MI455X (gfx1250) — compile-verified
//
#include <hip/hip_runtime.h>
#include <hip/hip_bf16.h>
#include <math.h>

// ---------------------------------------------------------------------------
// AGCRN cell on MI455X (gfx1250, wave32, WGP).
// GEMMs via V_WMMA_F32_16X16X4_F32; B operand of the big adj-GEMM staged into
// LDS with CDNA5 async global->LDS copies (ASYNCcnt) + double buffering.
//
// Sizes: N=4096, IN=64, QD=16, CI=129 (padded KP=144), DOUT=QD*IN=1024.
// ---------------------------------------------------------------------------

typedef __attribute__((ext_vector_type(2))) float v2f;
typedef __attribute__((ext_vector_type(8))) float v8f;

#define NN    4096
#define INSZ  64
#define QD    16
#define CI    129
#define KP    144          // CI padded to multiple of 16
#define DOUT  1024         // QD * INSZ

#define KSPLIT 8           // K-slices for adj GEMM (parallelism)
#define KBLK   (NN / KSPLIT)   // 512 K per block
#define KC     64              // K rows staged in LDS per chunk
#define NCH    (KBLK / KC)     // 8 chunks per block
#define BROWS  128             // rows per block = 8 waves * 16
#define CHUNK_FLOATS (KC * KP)         // 9216 floats = 36864 B
#define CHUNK_F4     (CHUNK_FLOATS/4)  // 2304 float4 transfers per chunk

// --- CDNA5 async global->LDS copy (tracked by ASYNCcnt) --------------------
__device__ __forceinline__ void async_copy_b128(unsigned lds_byte_off,
                                                unsigned long long gaddr)
{
    asm volatile("global_load_async_to_lds_b128 %0, %1, off"
                 :: "v"(lds_byte_off), "v"(gaddr)
                 : "memory");
}
__device__ __forceinline__ void wait_asynccnt_le9()
{   asm volatile("s_wait_asynccnt 0x9" ::: "memory"); }
__device__ __forceinline__ void wait_asynccnt_0()
{   asm volatile("s_wait_asynccnt 0x0" ::: "memory"); }

// ---------------------------------------------------------------------------
// GEMM1: partial[ks] = adj[:, ks-slice] @ XHp[ks-slice, :]
// Block: 256 threads (8 waves), 128 rows x 144 cols, K-slice of 512.
// Wave computes a 16x144 strip: 9 fp32 WMMA accumulators; A loaded once per
// K-step (amortized over 9 WMMAs); B read from LDS (async double-buffered).
// ---------------------------------------------------------------------------
__global__ __launch_bounds__(256) void gemm1_adj_kernel(
    float* __restrict__ P,            // [KSPLIT][NN][KP] partial sums
    const float* __restrict__ A,      // adj [NN][NN]
    const float* __restrict__ B)      // XHp [NN][KP]
{
    __shared__ float ldsB[2 * CHUNK_FLOATS];   // 73728 B double buffer

    const int tid  = threadIdx.x;
    const int wave = tid >> 5;
    const unsigned lane = tid & 31;
    const unsigned half = lane >> 4;    // K sub-offset 0 / 2
    const unsigned lid  = lane & 15;

    const int rowBase = blockIdx.x * BROWS + wave * 16;
    const int kBase   = blockIdx.y * KBLK;

    const float* __restrict__ Ap =
        A + (size_t)(rowBase + lid) * NN + kBase + half * 2;

    v8f zero = {};
    v8f acc[9];
#pragma unroll
    for (int t = 0; t < 9; ++t) acc[t] = zero;

    const unsigned ldsBase = (unsigned)(uintptr_t)&ldsB[0];
    const unsigned chunkBytes = CHUNK_FLOATS * 4;

    // Prologue: kick off chunk 0 into buffer 0.
    {
        unsigned long long g0 = (unsigned long long)(B + (size_t)kBase * KP);
#pragma unroll
        for (int j = 0; j < 9; ++j) {
            unsigned f4 = (unsigned)tid + j * 256u;   // 2304 float4s / 256 thr
            async_copy_b128(ldsBase + f4 * 16u, g0 + (unsigned long long)f4 * 16u);
        }
    }

    for (int c = 0; c < NCH; ++c) {
        if (c + 1 < NCH) {
            // Prefetch next chunk into the other buffer (its previous reader,
            // chunk c-1, finished before the trailing barrier of iter c-1).
            unsigned long long g =
                (unsigned long long)(B + (size_t)(kBase + (c + 1) * KC) * KP);
            unsigned dst = ldsBase + (((c + 1) & 1) ? chunkBytes : 0u);
#pragma unroll
            for (int j = 0; j < 9; ++j) {
                unsigned f4 = (unsigned)tid + j * 256u;
                async_copy_b128(dst + f4 * 16u, g + (unsigned long long)f4 * 16u);
            }
            wait_asynccnt_le9();   // first 9 (chunk c) done; next 9 in flight
        } else {
            wait_asynccnt_0();
        }
        __syncthreads();           // all waves' chunk-c data visible in LDS

        const float* __restrict__ buf = ldsB + ((c & 1) ? CHUNK_FLOATS : 0);
        const float* __restrict__ Ac  = Ap + c * KC;
#pragma unroll 2
        for (int kk = 0; kk < KC; kk += 4) {
            v2f a;
            a.x = Ac[kk];
            a.y = Ac[kk + 1];
            const float* __restrict__ brow = buf + (kk + half * 2) * KP + lid;
#pragma unroll
            for (int t = 0; t < 9; ++t) {
                v2f b;
                b.x = brow[t * 16];        // row k+half*2
                b.y = brow[KP + t * 16];   // row k+half*2+1
                acc[t] = __builtin_amdgcn_wmma_f32_16x16x4_f32(
                    false, a, false, b, (short)0, acc[t], false, false);
            }
        }
        __syncthreads();           // safe to overwrite this buffer next iter+1
    }

    float* __restrict__ Cp = P + (size_t)blockIdx.y * NN * KP
                               + (size_t)(rowBase + half * 8) * KP + lid;
#pragma unroll
    for (int t = 0; t < 9; ++t)
#pragma unroll
        for (int v = 0; v < 8; ++v)
            Cp[(size_t)v * KP + t * 16] = acc[t][v];
}

// Sum the KSPLIT partial slices (deterministic fp32 reduction).
__global__ void reduce_partials_kernel(const float* __restrict__ P,
                                       float* __restrict__ C)
{
    int i = blockIdx.x * blockDim.x + threadIdx.x;
    if (i >= NN * KP) return;
    float s = 0.0f;
#pragma unroll
    for (int k = 0; k < KSPLIT; ++k) s += P[(size_t)k * NN * KP + i];
    C[i] = s;
}

// ---------------------------------------------------------------------------
// Generic fp32 WMMA GEMM, NT column tiles per wave (A load amortized NT x).
// C[M,N] = A[M,K] @ B[K,N], row-major, M%16==0, N%(16*NT)==0, K%4==0.
// ---------------------------------------------------------------------------
template<int NT>
__global__ __launch_bounds__(32) void gemm_wmma_f32_nt(
    float* __restrict__ C, const float* __restrict__ A, const float* __restrict__ B,
    int K, int lda, int ldb, int ldc)
{
    const int tm  = blockIdx.x;
    const int tn0 = blockIdx.y * NT;
    const unsigned lane = threadIdx.x;
    const unsigned half = lane >> 4;
    const unsigned lid  = lane & 15;

    const float* __restrict__ Ap = A + (size_t)(tm * 16 + lid) * lda + half * 2;
    const float* __restrict__ Bp = B + (size_t)(half * 2) * ldb + tn0 * 16 + lid;

    v8f zero = {};
    v8f acc[NT];
#pragma unroll
    for (int t = 0; t < NT; ++t) acc[t] = zero;

#pragma unroll 2
    for (int k = 0; k < K; k += 4) {
        v2f a;
        a.x = Ap[k];
        a.y = Ap[k + 1];
        const float* __restrict__ bk = Bp + (size_t)k * ldb;
#pragma unroll
        for (int t = 0; t < NT; ++t) {
            v2f b;
            b.x = bk[t * 16];
            b.y = bk[(size_t)ldb + t * 16];
            acc[t] = __builtin_amdgcn_wmma_f32_16x16x4_f32(
                false, a, false, b, (short)0, acc[t], false, false);
        }
    }

    float* __restrict__ Cp = C + (size_t)(tm * 16 + half * 8) * ldc + tn0 * 16 + lid;
#pragma unroll
    for (int t = 0; t < NT; ++t)
#pragma unroll
        for (int v = 0; v < 8; ++v)
            Cp[(size_t)v * ldc + t * 16] = acc[t][v];
}

// --------------------------- packing / epilogues ---------------------------
__global__ void pack_xh_kernel(const float* __restrict__ x, const float* __restrict__ h,
                               float* __restrict__ XHp)
{
    int i = blockIdx.x * blockDim.x + threadIdx.x;
    if (i >= NN * KP) return;
    int n = i / KP, c = i - n * KP;
    float v = 0.0f;
    if (c < INSZ + 1)      v = x[(size_t)n * (INSZ + 1) + c];
    else if (c < CI)       v = h[(size_t)n * INSZ + (c - (INSZ + 1))];
    XHp[i] = v;
}

__global__ void pack_w_kernel(const float* __restrict__ W, float* __restrict__ Wf)
{
    int i = blockIdx.x * blockDim.x + threadIdx.x;
    if (i >= KP * DOUT) return;
    int k = i / DOUT, c = i - k * DOUT;
    int d = c >> 6, o = c & 63;
    Wf[i] = (k < CI) ? W[((size_t)d * CI + k) * INSZ + o] : 0.0f;
}

__global__ void gate_kernel(const float* __restrict__ q,
                            const float* __restrict__ Tr, const float* __restrict__ Tu,
                            const float* __restrict__ br, const float* __restrict__ bu,
                            const float* __restrict__ h,
                            float* __restrict__ h2, float* __restrict__ u_out)
{
    int i = blockIdx.x * blockDim.x + threadIdx.x;
    if (i >= NN * INSZ) return;
    int n = i >> 6, o = i & 63;
    const float* qn  = q  + (size_t)n * QD;
    const float* trn = Tr + (size_t)n * DOUT + o;
    const float* tun = Tu + (size_t)n * DOUT + o;
    float sr = 0.0f, su = 0.0f;
#pragma unroll
    for (int d = 0; d < QD; ++d) {
        float qd = qn[d];
        sr = fmaf(qd, trn[d * INSZ] + br[d * INSZ + o], sr);
        su = fmaf(qd, tun[d * INSZ] + bu[d * INSZ + o], su);
    }
    float r = 1.0f / (1.0f + __expf(-sr));
    float u = 1.0f / (1.0f + __expf(-su));
    h2[i]    = r * h[i];
    u_out[i] = u;
}

__global__ void final_kernel(const float* __restrict__ q, const float* __restrict__ Tc,
                             const float* __restrict__ bc, const float* __restrict__ h2,
                             const float* __restrict__ u, float* __restrict__ out)
{
    int i = blockIdx.x * blockDim.x + threadIdx.x;
    if (i >= NN * INSZ) return;
    int n = i >> 6, o = i & 63;
    const float* qn  = q  + (size_t)n * QD;
    const float* tcn = Tc + (size_t)n * DOUT + o;
    float sc = 0.0f;
#pragma unroll
    for (int d = 0; d < QD; ++d)
        sc = fmaf(qn[d], tcn[d * INSZ] + bc[d * INSZ + o], sc);
    float cand = tanhf(sc);
    float uu = u[i];
    out[i] = (1.0f - uu) * h2[i] + uu * cand;
}

// ------------------------------- launcher ----------------------------------
extern "C" void kernel_launch(void* const* d_in, const int* in_sizes, int n_in,
                              void* d_out, int out_size, void* d_ws, size_t ws_size,
                              hipStream_t stream)
{
    const float* x   = (const float*)d_in[0];
    const float* h   = (const float*)d_in[1];
    const float* q   = (const float*)d_in[2];
    const float* adj = (const float*)d_in[3];
    const float* W_r = (const float*)d_in[5];
    const float* b_r = (const float*)d_in[6];
    const float* W_u = (const float*)d_in[7];
    const float* b_u = (const float*)d_in[8];
    const float* W_c = (const float*)d_in[9];
    const float* b_c = (const float*)d_in[10];
    float* out = (float*)d_out;

    float* ws = (float*)d_ws;
    size_t off = 0;
    float* XHp   = ws + off; off += (size_t)NN * KP;
    float* comb  = ws + off; off += (size_t)NN * KP;
    float* XH2p  = ws + off; off += (size_t)NN * KP;
    float* Wrf   = ws + off; off += (size_t)KP * DOUT;
    float* Wuf   = ws + off; off += (size_t)KP * DOUT;
    float* Wcf   = ws + off; off += (size_t)KP * DOUT;
    float* Tr    = ws + off; off += (size_t)NN * DOUT;
    float* Tu    = ws + off; off += (size_t)NN * DOUT;
    float* h2    = ws + off; off += (size_t)NN * INSZ;
    float* uArr  = ws + off; off += (size_t)NN * INSZ;
    float* Pbuf  = ws + off; off += (size_t)KSPLIT * NN * KP;
    float* Tc    = Tr;   // Tr fully consumed by gate_kernel before GEMM2c

    const int TPB = 256;
    int nW  = KP * DOUT;
    int nXH = NN * KP;
    int nE  = NN * INSZ;

    // 1) Packing.
    pack_w_kernel<<<(nW + TPB - 1) / TPB, TPB, 0, stream>>>(W_r, Wrf);
    pack_w_kernel<<<(nW + TPB - 1) / TPB, TPB, 0, stream>>>(W_u, Wuf);
    pack_w_kernel<<<(nW + TPB - 1) / TPB, TPB, 0, stream>>>(W_c, Wcf);
    pack_xh_kernel<<<(nXH + TPB - 1) / TPB, TPB, 0, stream>>>(x, h, XHp);

    // 2) combined = adj @ XHp  (K split over 8 slices, LDS-async-staged B).
    gemm1_adj_kernel<<<dim3(NN / BROWS, KSPLIT), 256, 0, stream>>>(Pbuf, adj, XHp);
    reduce_partials_kernel<<<(nXH + TPB - 1) / TPB, TPB, 0, stream>>>(Pbuf, comb);

    // 3) Tr = combined @ Wrf ; Tu = combined @ Wuf   (M=4096,N=1024,K=144).
    gemm_wmma_f32_nt<4><<<dim3(NN / 16, DOUT / 64), 32, 0, stream>>>(
        Tr, comb, Wrf, KP, KP, DOUT, DOUT);
    gemm_wmma_f32_nt<4><<<dim3(NN / 16, DOUT / 64), 32, 0, stream>>>(
        Tu, comb, Wuf, KP, KP, DOUT, DOUT);

    // 4) Gates -> h2, u.
    gate_kernel<<<(nE + TPB - 1) / TPB, TPB, 0, stream>>>(q, Tr, Tu, b_r, b_u, h, h2, uArr);

    // 5) Tc = [x|h2] @ Wcf.
    pack_xh_kernel<<<(nXH + TPB - 1) / TPB, TPB, 0, stream>>>(x, h2, XH2p);
    gemm_wmma_f32_nt<4><<<dim3(NN / 16, DOUT / 64), 32, 0, stream>>>(
        Tc, XH2p, Wcf, KP, KP, DOUT, DOUT);

    // 6) Blend.
    final_kernel<<<(nE + TPB - 1) / TPB, TPB, 0, stream>>>(q, Tc, b_c, h2, uArr, out);
}